// SimpleMolecularRegressor_84301618086274
// MI455X (gfx1250) — compile-verified
//
#include <hip/hip_runtime.h>
#include <hip/hip_bf16.h>
#include <stdint.h>

// ===========================================================================
// GraphSAGE molecular regressor for MI455X (gfx1250, wave32).
// fp32 end-to-end. Dense node-feature GEMMs: V_WMMA_F32_16X16X4_F32 with
// weights staged block-wide into LDS by the Tensor Data Mover
// (tensor_load_to_lds + s_wait_tensorcnt), B fragments read via ds_load.
// The edge scatter (~1.8 GB of L2 atomic traffic) remains the roofline
// bottleneck; GEMM precision therefore stays full fp32.
// ===========================================================================

typedef __attribute__((ext_vector_type(2))) float v2f;
typedef __attribute__((ext_vector_type(8))) float v8f;

#define BN_EPS 1e-5f

#if __has_builtin(__builtin_amdgcn_tensor_load_to_lds) && \
    __has_builtin(__builtin_amdgcn_s_wait_tensorcnt)
#define HAVE_TDM 1
#endif

#ifdef HAVE_TDM
typedef unsigned int u32x4 __attribute__((ext_vector_type(4)));
typedef int          i32x4 __attribute__((ext_vector_type(4)));
typedef int          i32x8 __attribute__((ext_vector_type(8)));

// TDM 2-D tile load: whole [64 x Kpad] fp32 matrix, global -> LDS.
// D# layout per CDNA5 ISA (group0 128b, group1 256b).
// If ldsStride != Kpad we request LDS padding: pad_interval=5 (every 64
// DWORDs = one row for Kpad=64), pad_amount=1 (2 DWORDs) -> stride 66.
__device__ __forceinline__ void tdm_load_2d(const float* gsrc, void* ldsDst,
                                            int Kpad, int ldsStride) {
  const unsigned long long ga = (unsigned long long)(uintptr_t)gsrc;
  const unsigned int lds = (unsigned int)(uintptr_t)ldsDst;  // LDS aperture low bits
  const int pad_on = (ldsStride != Kpad) ? 1 : 0;
  u32x4 g0;
  g0[0] = 1u;                                   // count=1, is_restore=0, gather=0
  g0[1] = lds;                                  // lds_addr (bytes)
  g0[2] = (unsigned int)(ga & 0xFFFFFFFFu);     // global_addr[31:0]
  g0[3] = (unsigned int)((ga >> 32) & 0x1FFFFFFu) | (2u << 30); // addr[56:32]|type=2
  i32x8 g1;
  g1[0] = (2 << 16)                             // data_size = 4 bytes
        | (pad_on << 20)                        // pad_enable
        | (pad_on ? ((5 << 22) | (1 << 25)) : 0); // pad_interval=5, pad_amount=1
  g1[1] = (Kpad & 0xFFFF) << 16;                // tensor_dim0 lo16 (atomic addr 0)
  g1[2] = (64 << 16);                           // tensor_dim0 hi16=0 | tensor_dim1=64
  g1[3] = (Kpad << 16);                         // tensor_dim1 hi16=0 | tile_dim0=Kpad
  g1[4] = 64;                                   // tile_dim1=64, tile_dim2=0
  g1[5] = Kpad;                                 // tensor_dim0_stride lo32
  g1[6] = ((Kpad * 64) & 0xFFFF) << 16;         // stride0 hi=0 | stride1 lo16
  g1[7] = (Kpad * 64) >> 16;                    // stride1 hi
  i32x4 z4 = {0, 0, 0, 0};
#if __clang_major__ >= 23
  i32x8 z8 = {0, 0, 0, 0, 0, 0, 0, 0};
  __builtin_amdgcn_tensor_load_to_lds(g0, g1, z4, z4, z8, 0);
#else
  __builtin_amdgcn_tensor_load_to_lds(g0, g1, z4, z4, 0);
#endif
}
#endif  // HAVE_TDM

// ---------------------------------------------------------------------------
// Utility kernels
// ---------------------------------------------------------------------------
__global__ void fill_f32_kernel(float* __restrict__ p, float v, int n) {
  int i = blockIdx.x * blockDim.x + threadIdx.x;
  if (i < n) p[i] = v;
}

// out[r, 0:dinPad] = { in[r, 0:din], zeros }   (row-major)
__global__ void pad_copy_kernel(const float* __restrict__ in, float* __restrict__ out,
                                int rows, int din, int dinPad) {
  int i = blockIdx.x * blockDim.x + threadIdx.x;
  if (i >= rows * dinPad) return;
  int r = i / dinPad, f = i - r * dinPad;
  out[i] = (f < din) ? in[r * din + f] : 0.0f;
}

// ---------------------------------------------------------------------------
// Edge scatter:  agg[dst,f] += x[src,f];  deg[dst] += 1  (once per edge)
// Consecutive threads cover consecutive features of one edge -> coalesced
// gather from x[src] and coalesced atomics into agg[dst]. Padded feature
// columns hold zeros, so scattering them is harmless.
// ---------------------------------------------------------------------------
template <int DINPAD>
__global__ void scatter_accum_kernel(const int* __restrict__ ei,   // [2, nE]
                                     const float* __restrict__ x,  // [N, DINPAD]
                                     float* __restrict__ agg,      // [N, DINPAD]
                                     float* __restrict__ deg,      // [N]
                                     int nE) {
  int tid = blockIdx.x * blockDim.x + threadIdx.x;
  if (tid >= nE * DINPAD) return;
  int e = tid / DINPAD, f = tid - e * DINPAD;
  int s = ei[e];        // row 0 = src
  int d = ei[nE + e];   // row 1 = dst
  atomicAdd(&agg[d * DINPAD + f], x[s * DINPAD + f]);
  if (f == 0) atomicAdd(&deg[d], 1.0f);
}

template <int DINPAD>
__global__ void mean_div_kernel(float* __restrict__ agg, const float* __restrict__ deg,
                                int nNodes) {
  int tid = blockIdx.x * blockDim.x + threadIdx.x;
  if (tid >= nNodes * DINPAD) return;
  agg[tid] = agg[tid] / fmaxf(deg[tid / DINPAD], 1.0f);
}

// ---------------------------------------------------------------------------
// WMMA fp32 GEMM:  h[m,n] = sum_k A[m,k]*Wl[n,k] + bl[n] + sum_k X[m,k]*Wr[n,k]
// then BN + ReLU (+ optional residual  out = resid + 0.5*h).
// One wave per 16-row tile; 4 col tiles of 16 (HID=64) in 4 v8f accumulators.
// Weights live in LDS (staged by TDM), A rows stream from global.
//
// V_WMMA_F32_16X16X4_F32 operand layout (ISA 7.12.2, 32-bit A 16x4):
//   lane<16 : M=lane,    VGPR0=K(k0),   VGPR1=K(k0+1)
//   lane>=16: M=lane-16, VGPR0=K(k0+2), VGPR1=K(k0+3)
// B (4x16) mirrored: B[k][n] = W[n][k].
// C/D 16x16: VGPR r -> M = r + 8*(lane/16), N = lane%16.
// ---------------------------------------------------------------------------
__device__ __forceinline__ void wmma_pair_accum(const float* __restrict__ in,
                                                const float* __restrict__ wlds,
                                                int rowA, int l16, int half,
                                                int Kpad, int wStride,
                                                v8f& c0, v8f& c1, v8f& c2, v8f& c3) {
  const float* arow = in + (size_t)rowA * Kpad;
  for (int k0 = 0; k0 < Kpad; k0 += 4) {
    const int ka = k0 + 2 * half;
    v2f a;
    a.x = arow[ka];
    a.y = arow[ka + 1];
    const float* wp = wlds + l16 * wStride + ka;   // LDS (ds_load_b64)
    v2f b0, b1, b2, b3;
    b0.x = wp[0];               b0.y = wp[1];
    b1.x = wp[16 * wStride];    b1.y = wp[16 * wStride + 1];
    b2.x = wp[32 * wStride];    b2.y = wp[32 * wStride + 1];
    b3.x = wp[48 * wStride];    b3.y = wp[48 * wStride + 1];
    c0 = __builtin_amdgcn_wmma_f32_16x16x4_f32(false, a, false, b0, (short)0, c0, false, false);
    c1 = __builtin_amdgcn_wmma_f32_16x16x4_f32(false, a, false, b1, (short)0, c1, false, false);
    c2 = __builtin_amdgcn_wmma_f32_16x16x4_f32(false, a, false, b2, (short)0, c2, false, false);
    c3 = __builtin_amdgcn_wmma_f32_16x16x4_f32(false, a, false, b3, (short)0, c3, false, false);
  }
}

__global__ __launch_bounds__(256)
void sage_gemm_kernel(const float* __restrict__ A,    // agg  [M, Kpad]
                      const float* __restrict__ X,    // x    [M, Kpad]
                      const float* __restrict__ Wl,   // [64, Kpad]
                      const float* __restrict__ bl,   // [64]
                      const float* __restrict__ Wr,   // [64, Kpad]
                      const float* __restrict__ bng, const float* __restrict__ bnb,
                      const float* __restrict__ bnm, const float* __restrict__ bnv,
                      const float* __restrict__ resid, // [M, 64] or nullptr
                      float* __restrict__ out,         // [M, 64]
                      int M, int Kpad, int ldsStride) {
  // 2 weight matrices, 64 rows each, max LDS stride 66 floats -> 33 KB.
  __shared__ float shw[2 * 64 * 66];

  // ---- stage Wl, Wr into LDS (TDM if available, cooperative copy else) ----
#ifdef HAVE_TDM
  if (threadIdx.x == 0) {
    tdm_load_2d(Wl, shw, Kpad, ldsStride);
    tdm_load_2d(Wr, shw + 64 * ldsStride, Kpad, ldsStride);
    __builtin_amdgcn_s_wait_tensorcnt(0);
  }
#else
  for (int i = threadIdx.x; i < 2 * 64 * Kpad; i += 256) {
    int r = i / Kpad, c = i - r * Kpad;
    shw[r * ldsStride + c] = (r < 64) ? Wl[i] : Wr[(r - 64) * Kpad + c];
  }
#endif
  __syncthreads();

  const int wave  = threadIdx.x >> 5;
  const int lane  = threadIdx.x & 31;
  const int tileM = blockIdx.x * 8 + wave;
  if (tileM * 16 >= M) return;                 // wave-uniform -> EXEC stays full
  const int half = lane >> 4;
  const int l16  = lane & 15;
  const int rowA = tileM * 16 + l16;           // A row served by this lane

  v8f c0 = {}, c1 = {}, c2 = {}, c3 = {};
  wmma_pair_accum(A, shw,                  rowA, l16, half, Kpad, ldsStride, c0, c1, c2, c3);
  wmma_pair_accum(X, shw + 64 * ldsStride, rowA, l16, half, Kpad, ldsStride, c0, c1, c2, c3);

  v8f cc[4] = {c0, c1, c2, c3};
#pragma unroll
  for (int t = 0; t < 4; ++t) {
    const int col   = t * 16 + l16;
    const float scl = bng[col] * rsqrtf(bnv[col] + BN_EPS);
    const float bia = bl[col];
    const float bm  = bnm[col];
    const float bb  = bnb[col];
#pragma unroll
    for (int r = 0; r < 8; ++r) {
      const int row = tileM * 16 + r + half * 8;
      float v = cc[t][r] + bia;
      v = (v - bm) * scl + bb;      // BatchNorm (eval)
      v = fmaxf(v, 0.0f);           // ReLU
      if (resid) v = resid[row * 64 + col] + 0.5f * v;
      out[row * 64 + col] = v;
    }
  }
}

// ---------------------------------------------------------------------------
// Per-graph pooling. Node features are post-ReLU (>= 0), so max-pool is done
// with atomicMax on the raw uint bit pattern over a zero-initialized buffer
// (IEEE754 non-negative floats are monotone as unsigned ints; 0x0 == 0.0f).
// ---------------------------------------------------------------------------
__global__ void pool_scatter_kernel(const float* __restrict__ h,     // [N, 64]
                                    const int* __restrict__ batch,   // [N]
                                    float* __restrict__ gsum,        // [G, 64]
                                    unsigned int* __restrict__ gmax, // [G, 64]
                                    float* __restrict__ gcnt,        // [G]
                                    int nNodes) {
  int tid = blockIdx.x * blockDim.x + threadIdx.x;
  if (tid >= nNodes * 64) return;
  int i = tid >> 6, f = tid & 63;
  int g = batch[i];
  float v = h[tid];
  atomicAdd(&gsum[g * 64 + f], v);
  atomicMax(&gmax[g * 64 + f], __float_as_uint(v));
  if (f == 0) atomicAdd(&gcnt[g], 1.0f);
}

// ---------------------------------------------------------------------------
// Head: combined = BN([mean | max | adme]); 158->128 (BN,ReLU) -> 64 (BN,ReLU)
// -> 1. One block per graph, staged in LDS. ~29K MACs/graph — VALU is plenty.
// ---------------------------------------------------------------------------
__global__ __launch_bounds__(160)
void head_kernel(const float* __restrict__ gsum, const float* __restrict__ gmax,
                 const float* __restrict__ gcnt, const float* __restrict__ adme,
                 const float* __restrict__ cg, const float* __restrict__ cb,
                 const float* __restrict__ cm, const float* __restrict__ cv,
                 const float* __restrict__ w1, const float* __restrict__ b1,
                 const float* __restrict__ g1, const float* __restrict__ bb1,
                 const float* __restrict__ m1, const float* __restrict__ v1,
                 const float* __restrict__ w2, const float* __restrict__ b2,
                 const float* __restrict__ g2, const float* __restrict__ bb2,
                 const float* __restrict__ m2, const float* __restrict__ v2,
                 const float* __restrict__ w3, const float* __restrict__ b3,
                 float* __restrict__ out) {
  __shared__ float comb[158];
  __shared__ float h1[128];
  __shared__ float h2[64];
  const int g = blockIdx.x;
  const int t = threadIdx.x;

  if (t < 158) {
    float val;
    if (t < 64)       val = gsum[g * 64 + t] / fmaxf(gcnt[g], 1.0f);
    else if (t < 128) val = gmax[g * 64 + (t - 64)];
    else              val = adme[g * 30 + (t - 128)];
    comb[t] = (val - cm[t]) * (cg[t] * rsqrtf(cv[t] + BN_EPS)) + cb[t];
  }
  __syncthreads();
  if (t < 128) {
    float acc = b1[t];
    for (int k = 0; k < 158; ++k) acc += w1[t * 158 + k] * comb[k];
    acc = (acc - m1[t]) * (g1[t] * rsqrtf(v1[t] + BN_EPS)) + bb1[t];
    h1[t] = fmaxf(acc, 0.0f);
  }
  __syncthreads();
  if (t < 64) {
    float acc = b2[t];
    for (int k = 0; k < 128; ++k) acc += w2[t * 128 + k] * h1[k];
    acc = (acc - m2[t]) * (g2[t] * rsqrtf(v2[t] + BN_EPS)) + bb2[t];
    h2[t] = fmaxf(acc, 0.0f);
  }
  __syncthreads();
  if (t == 0) {
    float acc = b3[0];
    for (int k = 0; k < 64; ++k) acc += w3[k] * h2[k];
    out[g] = acc;
  }
}

// ---------------------------------------------------------------------------
// Launch
// ---------------------------------------------------------------------------
static inline int ceil_div(int a, int b) { return (a + b - 1) / b; }

extern "C" void kernel_launch(void* const* d_in, const int* in_sizes, int n_in,
                              void* d_out, int out_size, void* d_ws, size_t ws_size,
                              hipStream_t stream) {
  (void)n_in; (void)out_size; (void)ws_size;
  // -------- input pointer map --------
  // Assumed flattening: setup_inputs() insertion order for top-level entries
  // (x, edge_index, batch, adme, params), and JAX pytree order inside params
  // (dict keys sorted alphabetically, lists by index):
  //   params -> comb_norm{b,g,m,v}, head{bn1{b,g,m,v}, bn2{b,g,m,v},
  //             l1{W,b}, l2{W,b}, l3{W,b}}, layers[i]{bn{b,g,m,v},
  //             lin_l{W,b}, lin_r{W}}
  const float* x     = (const float*)d_in[0];   // [N, 37]
  const int*   ei    = (const int*)  d_in[1];   // [2, nE]
  const int*   batch = (const int*)  d_in[2];   // [N]
  const float* adme  = (const float*)d_in[3];   // [G, 30]
  const float* comb_b = (const float*)d_in[4];
  const float* comb_g = (const float*)d_in[5];
  const float* comb_m = (const float*)d_in[6];
  const float* comb_v = (const float*)d_in[7];
  const float* bn1_b = (const float*)d_in[8];
  const float* bn1_g = (const float*)d_in[9];
  const float* bn1_m = (const float*)d_in[10];
  const float* bn1_v = (const float*)d_in[11];
  const float* bn2_b = (const float*)d_in[12];
  const float* bn2_g = (const float*)d_in[13];
  const float* bn2_m = (const float*)d_in[14];
  const float* bn2_v = (const float*)d_in[15];
  const float* l1_W  = (const float*)d_in[16];  // [128,158]
  const float* l1_b  = (const float*)d_in[17];
  const float* l2_W  = (const float*)d_in[18];  // [64,128]
  const float* l2_b  = (const float*)d_in[19];
  const float* l3_W  = (const float*)d_in[20];  // [1,64]
  const float* l3_b  = (const float*)d_in[21];
  // layers[i]: bn{b,g,m,v}, lin_l.W, lin_l.b, lin_r.W   (7 leaves each)
  const float* Lbn_b[3], *Lbn_g[3], *Lbn_m[3], *Lbn_v[3];
  const float* Lwl[3], *Lbl[3], *Lwr[3];
  for (int i = 0; i < 3; ++i) {
    int base = 22 + 7 * i;
    Lbn_b[i] = (const float*)d_in[base + 0];
    Lbn_g[i] = (const float*)d_in[base + 1];
    Lbn_m[i] = (const float*)d_in[base + 2];
    Lbn_v[i] = (const float*)d_in[base + 3];
    Lwl[i]   = (const float*)d_in[base + 4];   // [64, din]
    Lbl[i]   = (const float*)d_in[base + 5];   // [64]
    Lwr[i]   = (const float*)d_in[base + 6];   // [64, din]
  }

  const int N  = in_sizes[2];          // 200000
  const int nE = in_sizes[1] / 2;      // 1200000
  const int G  = in_sizes[3] / 30;     // 8192
  const int DIN = 37, DINP = 40, HID = 64;
  const int STRIDE40 = 40;             // layer-0 LDS stride (no pow2 padding)
  const int STRIDE66 = 66;             // layer-1/2 LDS stride (TDM row pad)

  // -------- workspace layout --------
  char* ws = (char*)d_ws;
  size_t off = 0;
  auto take = [&](size_t bytes) { char* p = ws + off; off = (off + bytes + 255) & ~(size_t)255; return p; };
  float*        bufA = (float*)take((size_t)N * HID * 4);   // node features A
  float*        bufB = (float*)take((size_t)N * HID * 4);   // node features B (also x-padded)
  float*        agg  = (float*)take((size_t)N * HID * 4);   // neighbor aggregate
  float*        deg  = (float*)take((size_t)N * 4);
  float*        wlp  = (float*)take((size_t)HID * DINP * 4);
  float*        wrp  = (float*)take((size_t)HID * DINP * 4);
  float*        gsum = (float*)take((size_t)G * HID * 4);
  unsigned int* gmax = (unsigned int*)take((size_t)G * HID * 4);
  float*        gcnt = (float*)take((size_t)G * 4);

  const int B = 256;
  const int gemmGrid = ceil_div(N / 16, 8);   // 1 wave per 16-row tile, 8 waves/block

  // ======== Layer 0 (din=37, padded to 40) ========
  pad_copy_kernel<<<ceil_div(N * DINP, B), B, 0, stream>>>(x, bufB, N, DIN, DINP);
  pad_copy_kernel<<<ceil_div(HID * DINP, B), B, 0, stream>>>(Lwl[0], wlp, HID, DIN, DINP);
  pad_copy_kernel<<<ceil_div(HID * DINP, B), B, 0, stream>>>(Lwr[0], wrp, HID, DIN, DINP);
  fill_f32_kernel<<<ceil_div(N * DINP, B), B, 0, stream>>>(agg, 0.0f, N * DINP);
  fill_f32_kernel<<<ceil_div(N, B), B, 0, stream>>>(deg, 0.0f, N);
  scatter_accum_kernel<40><<<ceil_div(nE * DINP, B), B, 0, stream>>>(ei, bufB, agg, deg, nE);
  mean_div_kernel<40><<<ceil_div(N * DINP, B), B, 0, stream>>>(agg, deg, N);
  sage_gemm_kernel<<<gemmGrid, B, 0, stream>>>(agg, bufB, wlp, Lbl[0], wrp,
      Lbn_g[0], Lbn_b[0], Lbn_m[0], Lbn_v[0], nullptr, bufA, N, DINP, STRIDE40);

  // ======== Layer 1 (64 -> 64, residual x + 0.5*h) ========
  fill_f32_kernel<<<ceil_div(N * HID, B), B, 0, stream>>>(agg, 0.0f, N * HID);
  fill_f32_kernel<<<ceil_div(N, B), B, 0, stream>>>(deg, 0.0f, N);
  scatter_accum_kernel<64><<<ceil_div(nE * HID, B), B, 0, stream>>>(ei, bufA, agg, deg, nE);
  mean_div_kernel<64><<<ceil_div(N * HID, B), B, 0, stream>>>(agg, deg, N);
  sage_gemm_kernel<<<gemmGrid, B, 0, stream>>>(agg, bufA, Lwl[1], Lbl[1], Lwr[1],
      Lbn_g[1], Lbn_b[1], Lbn_m[1], Lbn_v[1], /*resid=*/bufA, bufB, N, HID, STRIDE66);

  // ======== Layer 2 (64 -> 64) ========
  fill_f32_kernel<<<ceil_div(N * HID, B), B, 0, stream>>>(agg, 0.0f, N * HID);
  fill_f32_kernel<<<ceil_div(N, B), B, 0, stream>>>(deg, 0.0f, N);
  scatter_accum_kernel<64><<<ceil_div(nE * HID, B), B, 0, stream>>>(ei, bufB, agg, deg, nE);
  mean_div_kernel<64><<<ceil_div(N * HID, B), B, 0, stream>>>(agg, deg, N);
  sage_gemm_kernel<<<gemmGrid, B, 0, stream>>>(agg, bufB, Lwl[2], Lbl[2], Lwr[2],
      Lbn_g[2], Lbn_b[2], Lbn_m[2], Lbn_v[2], nullptr, bufA, N, HID, STRIDE66);

  // ======== Readout ========
  fill_f32_kernel<<<ceil_div(G * HID, B), B, 0, stream>>>(gsum, 0.0f, G * HID);
  fill_f32_kernel<<<ceil_div(G * HID, B), B, 0, stream>>>((float*)gmax, 0.0f, G * HID);
  fill_f32_kernel<<<ceil_div(G, B), B, 0, stream>>>(gcnt, 0.0f, G);
  pool_scatter_kernel<<<ceil_div(N * HID, B), B, 0, stream>>>(bufA, batch, gsum, gmax, gcnt, N);

  head_kernel<<<G, 160, 0, stream>>>(gsum, (const float*)gmax, gcnt, adme,
      comb_g, comb_b, comb_m, comb_v,
      l1_W, l1_b, bn1_g, bn1_b, bn1_m, bn1_v,
      l2_W, l2_b, bn2_g, bn2_b, bn2_m, bn2_v,
      l3_W, l3_b, (float*)d_out);
}